// WeightedFilterPy_59012850647119
// MI455X (gfx1250) — compile-verified
//
#include <hip/hip_runtime.h>
#include <cstdint>

// Problem constants (match reference).
#define BATCH 4
#define CH 3
#define HH 1024
#define WW 1024
#define KK 7
#define RAD 3

// Tiling: 256 threads (8 waves), each thread -> 4 consecutive pixels x 3 channels.
#define TPX 16
#define TPY 16
#define TILE_W 64              // 16 threads * 4 px
#define TILE_H 16
#define LROWS (TILE_H + 2*RAD) // 22
#define LCOLS (TILE_W + 2*RAD) // 70
#define PITCH 72               // LDS row pitch (floats), multiple of 4 -> 16B-aligned b128 reads

typedef float v4f __attribute__((ext_vector_type(4)));
typedef float v2f __attribute__((ext_vector_type(2)));

// gfx1250 async global->LDS path (ASYNCcnt), guarded so both toolchains compile.
#if defined(__HIP_DEVICE_COMPILE__) && defined(__gfx1250__) && \
    __has_builtin(__builtin_amdgcn_global_load_async_to_lds_b32)
#define ASYNC_OK 1
typedef __attribute__((address_space(1))) int as1_int;
typedef __attribute__((address_space(3))) int as3_int;
#else
#define ASYNC_OK 0
#endif

__global__ __launch_bounds__(TPX * TPY)
void wfilter_kernel(const float* __restrict__ input,
                    const float* __restrict__ weight,
                    const float* __restrict__ bias,
                    float* __restrict__ out)
{
    __shared__ __align__(16) float smem[CH][LROWS][PITCH];

    const int tx = threadIdx.x;            // 0..15
    const int ty = threadIdx.y;            // 0..15
    const int tid = ty * TPX + tx;         // 0..255
    const int w0 = blockIdx.x * TILE_W;
    const int h0 = blockIdx.y * TILE_H;
    const int b  = blockIdx.z;

    // ---------------- Stage input tile (+3 halo) for 3 channels into LDS ----------------
    // LDS col lc maps to global col (w0 - 3 + lc); LDS row r maps to global row (h0 - 3 + r).
    const int NELEM = CH * LROWS * LCOLS;  // 4620
    for (int e = tid; e < NELEM; e += TPX * TPY) {
        int c   = e / (LROWS * LCOLS);
        int rem = e - c * (LROWS * LCOLS);
        int r   = rem / LCOLS;
        int col = rem - r * LCOLS;
        int gr = h0 - RAD + r;
        int gc = w0 - RAD + col;
        bool inb = (gr >= 0) & (gr < HH) & (gc >= 0) & (gc < WW);
        float* lp = &smem[c][r][col];
#if ASYNC_OK
        if (inb) {
            const float* gp = input + (((size_t)(b * CH + c) * HH + gr) * WW + gc);
            __builtin_amdgcn_global_load_async_to_lds_b32(
                (as1_int*)(uintptr_t)gp, (as3_int*)(uintptr_t)lp,
                /*offset=*/0, /*cpol=*/0);
        } else {
            *lp = 0.0f;   // zero padding
        }
#else
        float v = 0.0f;
        if (inb) v = input[((size_t)(b * CH + c) * HH + gr) * WW + gc];
        *lp = v;
#endif
    }
#if ASYNC_OK
#if __has_builtin(__builtin_amdgcn_s_wait_asynccnt)
    __builtin_amdgcn_s_wait_asynccnt(0);
#else
    asm volatile("s_wait_asynccnt 0" ::: "memory");
#endif
#endif
    __syncthreads();

    // ---------------- Compute: 4 px (along w) x 3 channels per thread ----------------
    const int h     = h0 + ty;
    const int wbase = w0 + 4 * tx;

    float acc0[4] = {0.f, 0.f, 0.f, 0.f};
    float acc1[4] = {0.f, 0.f, 0.f, 0.f};
    float acc2[4] = {0.f, 0.f, 0.f, 0.f};

    // Weight base for this (b, h, wbase); tap t adds t * HH * WW floats.
    const float* wp = weight + ((size_t)b * (KK * KK) * HH * WW + (size_t)h * WW + wbase);

#pragma unroll
    for (int ii = 0; ii < KK; ++ii) {
        // Sliding window: input cols needed = wbase + (p + jj - 3), p+jj in [0,9]
        //   -> LDS cols [4*tx .. 4*tx+9], base 16B-aligned.
        float rw0[10], rw1[10], rw2[10];
        {
            const float* p0 = &smem[0][ty + ii][4 * tx];
            const float* p1 = &smem[1][ty + ii][4 * tx];
            const float* p2 = &smem[2][ty + ii][4 * tx];
            v4f a0 = ((const v4f*)p0)[0], b0 = ((const v4f*)p0)[1];
            v2f c0 = *(const v2f*)(p0 + 8);
            v4f a1 = ((const v4f*)p1)[0], b1 = ((const v4f*)p1)[1];
            v2f c1 = *(const v2f*)(p1 + 8);
            v4f a2 = ((const v4f*)p2)[0], b2 = ((const v4f*)p2)[1];
            v2f c2 = *(const v2f*)(p2 + 8);
#pragma unroll
            for (int q = 0; q < 4; ++q) { rw0[q] = a0[q]; rw1[q] = a1[q]; rw2[q] = a2[q]; }
#pragma unroll
            for (int q = 0; q < 4; ++q) { rw0[4+q] = b0[q]; rw1[4+q] = b1[q]; rw2[4+q] = b2[q]; }
            rw0[8] = c0[0]; rw0[9] = c0[1];
            rw1[8] = c1[0]; rw1[9] = c1[1];
            rw2[8] = c2[0]; rw2[9] = c2[1];
        }

#pragma unroll
        for (int jj = 0; jj < KK; ++jj) {
            // Per-pixel weights for tap (ii,jj): contiguous 4 floats, used once -> NT.
            const size_t toff = (size_t)(ii * KK + jj) * (HH * WW);
            v4f wv = __builtin_nontemporal_load((const v4f*)(wp + toff));
#pragma unroll
            for (int p = 0; p < 4; ++p) {
                acc0[p] += rw0[jj + p] * wv[p];
                acc1[p] += rw1[jj + p] * wv[p];
                acc2[p] += rw2[jj + p] * wv[p];
            }
        }
    }

    // ---------------- Bias + NT store ----------------
    const float bz0 = bias[0], bz1 = bias[1], bz2 = bias[2];
    v4f o0, o1, o2;
#pragma unroll
    for (int p = 0; p < 4; ++p) {
        o0[p] = acc0[p] + bz0;
        o1[p] = acc1[p] + bz1;
        o2[p] = acc2[p] + bz2;
    }
    float* ob = out + ((size_t)b * CH * HH * WW + (size_t)h * WW + wbase);
    __builtin_nontemporal_store(o0, (v4f*)(ob + 0 * (size_t)HH * WW));
    __builtin_nontemporal_store(o1, (v4f*)(ob + 1 * (size_t)HH * WW));
    __builtin_nontemporal_store(o2, (v4f*)(ob + 2 * (size_t)HH * WW));
}

extern "C" void kernel_launch(void* const* d_in, const int* in_sizes, int n_in,
                              void* d_out, int out_size, void* d_ws, size_t ws_size,
                              hipStream_t stream) {
    const float* input  = (const float*)d_in[0];   // (4,3,1024,1024) f32
    const float* weight = (const float*)d_in[1];   // (4,49,1024,1024) f32
    const float* bias   = (const float*)d_in[2];   // (3,) f32
    float* out = (float*)d_out;                    // (4,3,1024,1024) f32

    dim3 grid(WW / TILE_W, HH / TILE_H, BATCH);    // (16, 64, 4)
    dim3 block(TPX, TPY);                          // 256 threads = 8 waves
    hipLaunchKernelGGL(wfilter_kernel, grid, block, 0, stream, input, weight, bias, out);
}